// BatchGaussianRasterizer_39505109189227
// MI455X (gfx1250) — compile-verified
//
#include <hip/hip_runtime.h>
#include <hip/hip_bf16.h>

typedef float v2f __attribute__((ext_vector_type(2)));
typedef float v8f __attribute__((ext_vector_type(8)));

#define PB 1024
#define BB 2
#define HH 128
#define WW 128
#define STR 1040   // padded LDS row stride (floats): keeps channel rows in distinct banks

// ---------------------------------------------------------------------------
// Phase 1: per-(batch, gaussian) preprocessing. 2048 threads total.
// ---------------------------------------------------------------------------
__global__ __launch_bounds__(256) void gs_preprocess(
    const float* __restrict__ means, const float* __restrict__ opac,
    const float* __restrict__ scales, const float* __restrict__ rots,
    const float* __restrict__ sh, const float* __restrict__ viewms,
    const float* __restrict__ projms, const float* __restrict__ camposes,
    const float* __restrict__ tanxs, const float* __restrict__ tanys,
    float* __restrict__ U, int* __restrict__ radii_out)
{
  int gid = blockIdx.x * blockDim.x + threadIdx.x;
  if (gid >= BB * PB) return;
  int b = gid / PB, p = gid % PB;
  const float* V  = viewms + b * 16;
  const float* PM = projms + b * 16;
  float tanx = tanxs[b], tany = tanys[b];
  float cpx = camposes[b*3+0], cpy = camposes[b*3+1], cpz = camposes[b*3+2];
  float mx = means[p*3+0], my = means[p*3+1], mz = means[p*3+2];

  // t = p_hom @ V (row-vector times row-major 4x4)
  float t0 = mx*V[0] + my*V[4] + mz*V[8]  + V[12];
  float t1 = mx*V[1] + my*V[5] + mz*V[9]  + V[13];
  float t2 = mx*V[2] + my*V[6] + mz*V[10] + V[14];
  // p_proj = p_hom @ PM
  float q0 = mx*PM[0] + my*PM[4] + mz*PM[8]  + PM[12];
  float q1 = mx*PM[1] + my*PM[5] + mz*PM[9]  + PM[13];
  float q3 = mx*PM[3] + my*PM[7] + mz*PM[11] + PM[15];
  float inw  = 1.0f / (q3 + 1e-7f);
  float ndc0 = q0 * inw, ndc1 = q1 * inw;

  float depth = t2;
  float fx = (float)WW / (2.0f * tanx), fy = (float)HH / (2.0f * tany);
  float tz = fmaxf(depth, 1e-6f);
  float limx = 1.3f * tanx, limy = 1.3f * tany;
  float txc = fminf(limx, fmaxf(-limx, t0 / tz)) * tz;
  float tyc = fminf(limy, fmaxf(-limy, t1 / tz)) * tz;
  float itz = 1.0f / tz;
  float J00 = fx * itz, J02 = -fx * txc * itz * itz;
  float J11 = fy * itz, J12 = -fy * tyc * itz * itz;
  // T2 = J * W,  W[k][j] = V[j][k]  =>  T2[i][j] = sum_k J[i][k]*V[j*4+k]
  float T00 = J00*V[0*4+0] + J02*V[0*4+2];
  float T01 = J00*V[1*4+0] + J02*V[1*4+2];
  float T02 = J00*V[2*4+0] + J02*V[2*4+2];
  float T10 = J11*V[0*4+1] + J12*V[0*4+2];
  float T11 = J11*V[1*4+1] + J12*V[1*4+2];
  float T12 = J11*V[2*4+1] + J12*V[2*4+2];

  // quaternion -> rotation
  float qr = rots[p*4+0], qx = rots[p*4+1], qy = rots[p*4+2], qz = rots[p*4+3];
  float qn = rsqrtf(qr*qr + qx*qx + qy*qy + qz*qz);
  qr *= qn; qx *= qn; qy *= qn; qz *= qn;
  float R00 = 1.f-2.f*(qy*qy+qz*qz), R01 = 2.f*(qx*qy-qr*qz), R02 = 2.f*(qx*qz+qr*qy);
  float R10 = 2.f*(qx*qy+qr*qz), R11 = 1.f-2.f*(qx*qx+qz*qz), R12 = 2.f*(qy*qz-qr*qx);
  float R20 = 2.f*(qx*qz-qr*qy), R21 = 2.f*(qy*qz+qr*qx), R22 = 1.f-2.f*(qx*qx+qy*qy);
  float s0 = scales[p*3+0]; s0 *= s0;
  float s1 = scales[p*3+1]; s1 *= s1;
  float s2 = scales[p*3+2]; s2 *= s2;
  // Sigma = R diag(s2) R^T
  float S00 = R00*R00*s0 + R01*R01*s1 + R02*R02*s2;
  float S01 = R00*R10*s0 + R01*R11*s1 + R02*R12*s2;
  float S02 = R00*R20*s0 + R01*R21*s1 + R02*R22*s2;
  float S11 = R10*R10*s0 + R11*R11*s1 + R12*R12*s2;
  float S12 = R10*R20*s0 + R11*R21*s1 + R12*R22*s2;
  float S22 = R20*R20*s0 + R21*R21*s1 + R22*R22*s2;
  // cov2d = T2 Sigma T2^T
  float M00 = T00*S00 + T01*S01 + T02*S02;
  float M01 = T00*S01 + T01*S11 + T02*S12;
  float M02 = T00*S02 + T01*S12 + T02*S22;
  float M10 = T10*S00 + T11*S01 + T12*S02;
  float M11 = T10*S01 + T11*S11 + T12*S12;
  float M12 = T10*S02 + T11*S12 + T12*S22;
  float cov00 = M00*T00 + M01*T01 + M02*T02;
  float cov01 = M00*T10 + M01*T11 + M02*T12;
  float cov11 = M10*T10 + M11*T11 + M12*T12;

  float a  = cov00 + 0.3f, cc = cov11 + 0.3f, bb = cov01;
  float det = a*cc - bb*bb;
  bool valid = (det > 0.0f) && (depth > 0.2f);
  float idet = valid ? (1.0f / det) : 0.0f;     // conic zeroed when invalid
  float conA = cc * idet, conB = -bb * idet, conC = a * idet;
  float mid = 0.5f * (a + cc);
  float lam = mid + sqrtf(fmaxf(0.1f, mid*mid - det));
  int rad = valid ? (int)ceilf(3.0f * sqrtf(lam)) : 0;
  float px = ((ndc0 + 1.0f) * WW - 1.0f) * 0.5f;
  float py = ((ndc1 + 1.0f) * HH - 1.0f) * 0.5f;

  // SH degree 3 evaluation
  float dx = mx - cpx, dy = my - cpy, dz = mz - cpz;
  float dn = rsqrtf(dx*dx + dy*dy + dz*dz);
  float x = dx*dn, y = dy*dn, z = dz*dn;
  float xx = x*x, yy = y*y, zz = z*z, xy = x*y, yz = y*z, xz = x*z;
  float bas[16];
  bas[0]  = 0.28209479177387814f;
  bas[1]  = -0.4886025119029199f * y;
  bas[2]  =  0.4886025119029199f * z;
  bas[3]  = -0.4886025119029199f * x;
  bas[4]  =  1.0925484305920792f * xy;
  bas[5]  = -1.0925484305920792f * yz;
  bas[6]  =  0.31539156525252005f * (2.0f*zz - xx - yy);
  bas[7]  = -1.0925484305920792f * xz;
  bas[8]  =  0.5462742152960396f * (xx - yy);
  bas[9]  = -0.5900435899266435f * y * (3.0f*xx - yy);
  bas[10] =  2.890611442640554f  * xy * z;
  bas[11] = -0.4570457994644658f * y * (4.0f*zz - xx - yy);
  bas[12] =  0.3731763325901154f * z * (2.0f*zz - 3.0f*xx - 3.0f*yy);
  bas[13] = -0.4570457994644658f * x * (4.0f*zz - xx - yy);
  bas[14] =  1.445305721320277f  * z * (xx - yy);
  bas[15] = -0.5900435899266435f * x * (xx - 3.0f*yy);
  float col[3];
  #pragma unroll
  for (int c = 0; c < 3; ++c) {
    float acc = 0.0f;
    #pragma unroll
    for (int j = 0; j < 16; ++j) acc += bas[j] * sh[p*48 + j*3 + c];
    col[c] = fmaxf(acc + 0.5f, 0.0f);
  }
  float invd = 1.0f / tz;
  float opst = valid ? opac[p] : 0.0f;

  float* Ub = U + b * 12 * PB;
  Ub[0*PB + p]  = depth;
  Ub[1*PB + p]  = px;    Ub[2*PB + p]  = py;
  Ub[3*PB + p]  = conA;  Ub[4*PB + p]  = conB;  Ub[5*PB + p] = conC;
  Ub[6*PB + p]  = opst;
  Ub[7*PB + p]  = col[0]; Ub[8*PB + p] = col[1]; Ub[9*PB + p] = col[2];
  Ub[10*PB + p] = invd;
  radii_out[b*PB + p] = rad;
}

// ---------------------------------------------------------------------------
// Phase 2: LDS bitonic sort by depth (index tiebreak == stable argsort),
// then gather the 10 rasterizer fields into sorted SoA order.
// ---------------------------------------------------------------------------
__global__ __launch_bounds__(1024) void gs_sort(const float* __restrict__ U,
                                                float* __restrict__ S)
{
  __shared__ float key[PB];
  __shared__ int   idx[PB];
  int b = blockIdx.x, t = threadIdx.x;
  const float* Ub = U + b * 12 * PB;
  key[t] = Ub[t];
  idx[t] = t;
  __syncthreads();
  for (int k = 2; k <= PB; k <<= 1) {
    for (int j = k >> 1; j > 0; j >>= 1) {
      int ixj = t ^ j;
      if (ixj > t) {
        bool up = ((t & k) == 0);
        float ka = key[t], kb = key[ixj];
        int   ia = idx[t], ib = idx[ixj];
        bool agtb = (ka > kb) || (ka == kb && ia > ib);
        if (agtb == up) { key[t] = kb; key[ixj] = ka; idx[t] = ib; idx[ixj] = ia; }
      }
      __syncthreads();
    }
  }
  int src = idx[t];
  float* Sb = S + b * 10 * PB;
  #pragma unroll
  for (int f = 0; f < 10; ++f) Sb[f*PB + t] = Ub[(f+1)*PB + src];
}

// ---------------------------------------------------------------------------
// Phase 3: rasterize. One wave = 32 pixels (one lane owns one pixel and its
// serial transmittance chain). Per 4-gaussian chunk: alphas on VALU/TRANS,
// 4-channel accumulation on the matrix pipe via V_WMMA_F32_16X16X4_F32
// (D[16ch x 16px] += A[16ch x 4g] * B[4g x 16px]), co-executing with VALU.
// A-matrix is fetched branch-free: lane channel m<4 gathers from its channel
// row; m>=4 lanes gather from a zeroed 11th row. Row stride padded to 1040
// floats so the 4 channel rows map to distinct LDS banks.
// Block = 256 threads = 8 waves = 2 image rows.
// ---------------------------------------------------------------------------
__global__ __launch_bounds__(256) void gs_raster(const float* __restrict__ S,
                                                 const float* __restrict__ bgp,
                                                 float* __restrict__ out)
{
  __shared__ float lds[11 * STR];  // rows: px,py,cA,cB,cC,op,r,g,b,invd,zeros
  int b  = blockIdx.x / (HH / 2);
  int rp = blockIdx.x % (HH / 2);
  const float* Sb = S + b * 10 * PB;
  #pragma unroll
  for (int f = 0; f < 10; ++f)
    for (int i = threadIdx.x; i < PB; i += 256) lds[f*STR + i] = Sb[f*PB + i];
  for (int i = threadIdx.x; i < PB; i += 256) lds[10*STR + i] = 0.0f;
  __syncthreads();

  int wv = threadIdx.x >> 5;
  int L  = threadIdx.x & 31;
  int y  = rp * 2 + (wv >> 2);
  int xbase = (wv & 3) * 32;
  float fpx = (float)(xbase + L);
  float fpy = (float)y;

  int  m  = L & 15;
  bool lo = (L < 16);
  int  gofs = lo ? 0 : 2;
  // channel-row base for the branch-free A gather (r,g,b,invd rows or zero row)
  int cb = (m < 4) ? (6 + m) * STR : 10 * STR;

  float Tt = 1.0f;
  v8f acc1 = {0.f,0.f,0.f,0.f,0.f,0.f,0.f,0.f};
  v8f acc2 = {0.f,0.f,0.f,0.f,0.f,0.f,0.f,0.f};

  for (int c0 = 0; c0 < PB; c0 += 4) {
    float wk[4];
    #pragma unroll
    for (int k = 0; k < 4; ++k) {
      int g = c0 + k;
      float px = lds[g],         py = lds[STR + g];
      float cA = lds[2*STR + g], cB = lds[3*STR + g], cC = lds[4*STR + g];
      float op = lds[5*STR + g];
      float ddx = fpx - px, ddy = fpy - py;
      float pw = -0.5f * (cA*ddx*ddx + cC*ddy*ddy) - cB*ddx*ddy;
      float al = fminf(0.99f, op * __expf(pw));
      al = ((pw <= 0.0f) && (al >= (1.0f/255.0f))) ? al : 0.0f;
      wk[k] = Tt * al;
      Tt = Tt * (1.0f - al);
    }
    // A matrix: channel m of gaussians (c0+gofs, c0+gofs+1), branch-free gather
    int a0 = cb + c0 + gofs;
    float A0 = lds[a0];
    float A1 = lds[a0 + 1];
    v2f Av = {A0, A1};
    // B matrices for the two 16-pixel halves (rows K striped across lane halves)
    float s2a = __shfl(wk[2], m);
    float s3a = __shfl(wk[3], m);
    float s0b = __shfl(wk[0], m + 16);
    float s1b = __shfl(wk[1], m + 16);
    v2f B1 = { lo ? wk[0] : s2a, lo ? wk[1] : s3a };
    v2f B2 = { lo ? s0b : wk[2], lo ? s1b : wk[3] };
    acc1 = __builtin_amdgcn_wmma_f32_16x16x4_f32(false, Av, false, B1,
                                                 (short)0, acc1, false, false);
    acc2 = __builtin_amdgcn_wmma_f32_16x16x4_f32(false, Av, false, B2,
                                                 (short)0, acc2, false, false);
  }

  float bg0 = bgp[0], bg1 = bgp[1], bg2 = bgp[2];
  float Toth = __shfl(Tt, L + 16);   // T_final of pixel xbase+16+L (for lanes<16)
  if (lo) {
    int x1 = xbase + L, x2 = xbase + 16 + L;
    int cbase = b * 3 * HH * WW + y * WW;          // colors: (B,3,H,W)
    int ibase = BB*3*HH*WW + b * HH * WW + y * WW; // invdepth after colors
    out[cbase + x1]               = acc1[0] + bg0 * Tt;
    out[cbase + HH*WW + x1]       = acc1[1] + bg1 * Tt;
    out[cbase + 2*HH*WW + x1]     = acc1[2] + bg2 * Tt;
    out[ibase + x1]               = acc1[3];
    out[cbase + x2]               = acc2[0] + bg0 * Toth;
    out[cbase + HH*WW + x2]       = acc2[1] + bg1 * Toth;
    out[cbase + 2*HH*WW + x2]     = acc2[2] + bg2 * Toth;
    out[ibase + x2]               = acc2[3];
  }
}

// ---------------------------------------------------------------------------
extern "C" void kernel_launch(void* const* d_in, const int* in_sizes, int n_in,
                              void* d_out, int out_size, void* d_ws, size_t ws_size,
                              hipStream_t stream) {
  const float* means  = (const float*)d_in[0];
  const float* opac   = (const float*)d_in[1];
  const float* scales = (const float*)d_in[2];
  const float* rots   = (const float*)d_in[3];
  const float* sh     = (const float*)d_in[4];
  const float* bg     = (const float*)d_in[5];
  const float* viewms = (const float*)d_in[6];
  const float* projms = (const float*)d_in[7];
  const float* cps    = (const float*)d_in[8];
  const float* tanxs  = (const float*)d_in[9];
  const float* tanys  = (const float*)d_in[10];

  float* out = (float*)d_out;
  float* U = (float*)d_ws;             // unsorted: B * 12 * P floats
  float* S = U + BB * 12 * PB;         // sorted:   B * 10 * P floats
  int* radii = (int*)out + (BB*3*HH*WW + BB*HH*WW);  // tail of d_out, int32

  gs_preprocess<<<(BB*PB + 255)/256, 256, 0, stream>>>(
      means, opac, scales, rots, sh, viewms, projms, cps, tanxs, tanys,
      U, radii);
  gs_sort<<<BB, PB, 0, stream>>>(U, S);
  gs_raster<<<BB * (HH/2), 256, 0, stream>>>(S, bg, out);
}